// LightHeteroCF_23570780520850
// MI455X (gfx1250) — compile-verified
//
#include <hip/hip_runtime.h>
#include <cstdint>
#include <cstddef>

#define WAVE   32
#define WPB    8                 // waves per block in scatter kernel
#define SBLOCK (WAVE * WPB)

#if __has_builtin(__builtin_amdgcn_global_load_async_to_lds_b64)
#define HAVE_ASYNC_LDS 1
#endif

#ifdef HAVE_ASYNC_LDS
typedef int   v2i __attribute__((ext_vector_type(2)));
typedef float v2f __attribute__((ext_vector_type(2)));
typedef __attribute__((address_space(1))) v2i as1_v2i;
typedef __attribute__((address_space(3))) v2i as3_v2i;
typedef __attribute__((address_space(3))) v2f as3_v2f;
__device__ __forceinline__ void async_ld_b64(const float* g, float* l) {
  // global -> LDS async copy, 8 bytes per lane (ASYNCcnt-tracked)
  __builtin_amdgcn_global_load_async_to_lds_b64((as1_v2i*)g, (as3_v2i*)l, 0, 0);
}
// Explicit LDS-addrspace read so it lowers to ds_load_b64 (not a flat load).
__device__ __forceinline__ v2f lds_read_b64(float* l) {
  return *(as3_v2f*)l;
}
// SOPP wait on ASYNCcnt; "memory" clobber orders the LDS reads after the wait.
#define ASYNC_WAIT_LE_1() asm volatile("s_wait_asynccnt 0x1" ::: "memory")
#define ASYNC_WAIT_LE_0() asm volatile("s_wait_asynccnt 0x0" ::: "memory")
#endif

// ---------------- elementwise helpers ----------------

__global__ void zero4_kernel(float4* __restrict__ p, long long n4) {
  long long i = (long long)blockIdx.x * blockDim.x + threadIdx.x;
  long long stride = (long long)gridDim.x * blockDim.x;
  float4 z; z.x = 0.f; z.y = 0.f; z.z = 0.f; z.w = 0.f;
  for (; i < n4; i += stride) p[i] = z;
}

__global__ void accum4_kernel(float4* __restrict__ out, const float4* __restrict__ in,
                              long long n4) {
  long long i = (long long)blockIdx.x * blockDim.x + threadIdx.x;
  long long stride = (long long)gridDim.x * blockDim.x;
  for (; i < n4; i += stride) {
    float4 a = out[i];
    float4 b = in[i];
    a.x += b.x; a.y += b.y; a.z += b.z; a.w += b.w;
    out[i] = a;
  }
}

__global__ void scale4_kernel(float4* __restrict__ out, long long n4, float s) {
  long long i = (long long)blockIdx.x * blockDim.x + threadIdx.x;
  long long stride = (long long)gridDim.x * blockDim.x;
  for (; i < n4; i += stride) {
    float4 a = out[i];
    a.x *= s; a.y *= s; a.z *= s; a.w *= s;
    out[i] = a;
  }
}

// ---------------- degree kernels ----------------

__global__ void count_kernel(const int* __restrict__ src, float* __restrict__ deg, int n) {
  int i = blockIdx.x * blockDim.x + threadIdx.x;
  int stride = gridDim.x * blockDim.x;
  for (; i < n; i += stride) atomicAdd(&deg[src[i]], 1.0f);
}

// deg -> 0.5 / max(deg, 1)   (0.5 folds the "average of two edge sets")
__global__ void invdeg_kernel(float* __restrict__ d, long long n) {
  long long i = (long long)blockIdx.x * blockDim.x + threadIdx.x;
  long long stride = (long long)gridDim.x * blockDim.x;
  for (; i < n; i += stride) d[i] = 0.5f / fmaxf(d[i], 1.0f);
}

// ---------------- core scatter: agg[src] += xdst[dst] * invdeg[src] ----------------
// One wave per edge; lane carries 2 consecutive floats of the 64-wide row.
// Gather is staged through LDS with gfx1250 async global->LDS copies, double-buffered
// so the next edge's row is in flight while the current one feeds the atomic stream.

__global__ __launch_bounds__(SBLOCK)
void scatter_kernel(const int* __restrict__ srcIdx, const int* __restrict__ dstIdx,
                    const float* __restrict__ xdst, const float* __restrict__ invdeg,
                    float* __restrict__ agg, int nEdges) {
  const int lane = threadIdx.x & (WAVE - 1);
  const int wv   = threadIdx.x / WAVE;
  const long long wid     = (long long)blockIdx.x * WPB + wv;
  const long long wstride = (long long)gridDim.x * WPB;

#ifdef HAVE_ASYNC_LDS
  __shared__ float stage[WPB][2][64];
  long long e = wid;
  if (e >= nEdges) return;                       // wave-uniform exit
  int s0 = __builtin_amdgcn_readfirstlane(srcIdx[e]);
  int d0 = __builtin_amdgcn_readfirstlane(dstIdx[e]);
  int cur = 0;
  async_ld_b64(xdst + (size_t)d0 * 64 + lane * 2, &stage[wv][0][lane * 2]);
  for (; e < nEdges; e += wstride) {
    long long en = e + wstride;
    int s1 = 0, d1 = 0;
    if (en < nEdges) {                           // issue next tile, then wait for current
      s1 = __builtin_amdgcn_readfirstlane(srcIdx[en]);
      d1 = __builtin_amdgcn_readfirstlane(dstIdx[en]);
      async_ld_b64(xdst + (size_t)d1 * 64 + lane * 2, &stage[wv][cur ^ 1][lane * 2]);
      ASYNC_WAIT_LE_1();
    } else {
      ASYNC_WAIT_LE_0();
    }
    float w = invdeg[s0];
    v2f v = lds_read_b64(&stage[wv][cur][lane * 2]);   // ds_load_b64
    float* dp = agg + (size_t)s0 * 64 + lane * 2;
    atomicAdd(dp,     v.x * w);
    atomicAdd(dp + 1, v.y * w);
    s0 = s1; d0 = d1; cur ^= 1;
    (void)d0;
  }
#else
  for (long long e = wid; e < nEdges; e += wstride) {
    int s0 = __builtin_amdgcn_readfirstlane(srcIdx[e]);
    int d0 = __builtin_amdgcn_readfirstlane(dstIdx[e]);
    long long en = e + wstride;
    if (en < nEdges) {
      int dn = __builtin_amdgcn_readfirstlane(dstIdx[en]);
      __builtin_prefetch(xdst + (size_t)dn * 64 + lane * 2, 0, 0);  // global_prefetch_b8
    }
    const float2 v = *(const float2*)(xdst + (size_t)d0 * 64 + lane * 2);
    float w = invdeg[s0];
    float* dp = agg + (size_t)s0 * 64 + lane * 2;
    atomicAdd(dp,     v.x * w);
    atomicAdd(dp + 1, v.y * w);
  }
#endif
}

// ---------------- host orchestration ----------------

static inline unsigned blocks_for(long long n, int bs) {
  long long b = (n + bs - 1) / bs;
  if (b > 2147483647LL) b = 2147483647LL;
  if (b < 1) b = 1;
  return (unsigned)b;
}

extern "C" void kernel_launch(void* const* d_in, const int* in_sizes, int n_in,
                              void* d_out, int out_size, void* d_ws, size_t ws_size,
                              hipStream_t stream) {
  const float* user_emb  = (const float*)d_in[0];
  const float* paper_emb = (const float*)d_in[1];
  const int*   er = (const int*)d_in[2];   // [2,E]: row0 user ids, row1 paper ids
  const int*   ew = (const int*)d_in[3];
  const int*   rr = (const int*)d_in[4];   // [2,E]: row0 paper ids, row1 user ids
  const int*   rw = (const int*)d_in[5];

  const long long NU = (long long)in_sizes[0] / 64;
  const long long NP = (long long)in_sizes[1] / 64;
  const int E = in_sizes[2] / 2;

  // workspace layout (floats)
  float* bufU  = (float*)d_ws;             // [NU*64] current user state
  float* bufP  = bufU  + NU * 64;          // [NP*64] current paper state
  float* invER = bufP  + NP * 64;          // [NU]
  float* invEW = invER + NU;               // [NU]
  float* invRR = invEW + NU;               // [NP]
  float* invRW = invRR + NP;               // [NP]

  float* outU = (float*)d_out;             // running sum over layers
  float* outP = outU + NU * 64;

  const long long invTotal = 2 * NU + 2 * NP;
  const long long nU4 = NU * 16;           // NU*64 / 4
  const long long nP4 = NP * 16;

  // 1) per-edge-set inverse degrees (constant across layers)
  zero4_kernel<<<blocks_for(invTotal / 4, 256), 256, 0, stream>>>((float4*)invER, invTotal / 4);
  count_kernel<<<blocks_for(E, 256), 256, 0, stream>>>(er, invER, E);
  count_kernel<<<blocks_for(E, 256), 256, 0, stream>>>(ew, invEW, E);
  count_kernel<<<blocks_for(E, 256), 256, 0, stream>>>(rr, invRR, E);
  count_kernel<<<blocks_for(E, 256), 256, 0, stream>>>(rw, invRW, E);
  invdeg_kernel<<<blocks_for(invTotal, 256), 256, 0, stream>>>(invER, invTotal);

  // 2) init: layer-0 state & layer-0 term of the mean
  (void)hipMemcpyAsync(bufP, paper_emb, (size_t)NP * 64 * sizeof(float),
                       hipMemcpyDeviceToDevice, stream);
  (void)hipMemcpyAsync(outU, user_emb, (size_t)NU * 64 * sizeof(float),
                       hipMemcpyDeviceToDevice, stream);
  (void)hipMemcpyAsync(outP, paper_emb, (size_t)NP * 64 * sizeof(float),
                       hipMemcpyDeviceToDevice, stream);

  const unsigned sblocks = blocks_for((long long)(E + WPB - 1) / WPB, 1) > 4096u
                           ? 4096u : blocks_for((long long)(E + WPB - 1) / WPB, 1);

  // 3) three propagation layers
  for (int layer = 0; layer < 3; ++layer) {
    // x_u = sum_er x_p[dst]*(0.5/degER[src]) + sum_ew x_p[dst]*(0.5/degEW[src])
    zero4_kernel<<<blocks_for(nU4, 256), 256, 0, stream>>>((float4*)bufU, nU4);
    scatter_kernel<<<sblocks, SBLOCK, 0, stream>>>(er, er + E, bufP, invER, bufU, E);
    scatter_kernel<<<sblocks, SBLOCK, 0, stream>>>(ew, ew + E, bufP, invEW, bufU, E);
    accum4_kernel<<<blocks_for(nU4, 256), 256, 0, stream>>>((float4*)outU, (const float4*)bufU, nU4);

    // x_p = sum_rr x_u[dst]*(0.5/degRR[src]) + sum_rw x_u[dst]*(0.5/degRW[src])
    zero4_kernel<<<blocks_for(nP4, 256), 256, 0, stream>>>((float4*)bufP, nP4);
    scatter_kernel<<<sblocks, SBLOCK, 0, stream>>>(rr, rr + E, bufU, invRR, bufP, E);
    scatter_kernel<<<sblocks, SBLOCK, 0, stream>>>(rw, rw + E, bufU, invRW, bufP, E);
    accum4_kernel<<<blocks_for(nP4, 256), 256, 0, stream>>>((float4*)outP, (const float4*)bufP, nP4);
  }

  // 4) mean over 4 states (initial + 3 layers)
  scale4_kernel<<<blocks_for(nU4 + nP4, 256), 256, 0, stream>>>((float4*)outU, nU4 + nP4, 0.25f);
}